// ContextVector_48833778156098
// MI455X (gfx1250) — compile-verified
//
#include <hip/hip_runtime.h>
#include <hip/hip_bf16.h>

typedef __attribute__((ext_vector_type(2))) float v2f;
typedef __attribute__((ext_vector_type(4))) float v4f;
typedef __attribute__((ext_vector_type(8))) float v8f;
typedef __attribute__((ext_vector_type(4))) int   v4i;

#define AS1 __attribute__((address_space(1)))
#define AS3 __attribute__((address_space(3)))

// Problem constants (from reference setup_inputs)
#define BATCH   8
#define T_ENC   512
#define T_DEC   256
#define D_ENC   256
#define D_DEC   512

// ---------------------------------------------------------------------------
// CDNA5 feature shims (compile-safe: fall back if builtin absent)
// ---------------------------------------------------------------------------
__device__ __forceinline__ float fast_tanh(float x) {
#if __has_builtin(__builtin_amdgcn_tanhf)
    return __builtin_amdgcn_tanhf(x);      // v_tanh_f32 (single TRANS op)
#else
    return tanhf(x);
#endif
}

#if __has_builtin(__builtin_amdgcn_global_load_async_to_lds_b128) && \
    __has_builtin(__builtin_amdgcn_s_wait_asynccnt)
#define USE_ASYNC_LDS 1
#else
#define USE_ASYNC_LDS 0
#endif

// Copy 16 bytes global -> LDS
__device__ __forceinline__ void copy16_to_lds(const float* __restrict__ src,
                                              float* __restrict__ dst) {
#if USE_ASYNC_LDS
    // prototype (from clang diagnostic): (AS1 int4*, AS3 int4*, int, int)
    __builtin_amdgcn_global_load_async_to_lds_b128(
        (AS1 v4i*)src, (AS3 v4i*)dst, /*offset=*/0, /*cpol=*/0);
#else
    *(v4f*)dst = *(const v4f*)src;
#endif
}

__device__ __forceinline__ void wait_async_lds() {
#if USE_ASYNC_LDS
    __builtin_amdgcn_s_wait_asynccnt(0);
#endif
}

// ---------------------------------------------------------------------------
// Strided-batched fp32 GEMM using V_WMMA_F32_16X16X4_F32.
// C[M,N] = A[M,K] * B[K,N] (both row-major, contiguous). N,K compile-time so
// all addressing folds to shifts/immediates. One wave per 16x16 C tile.
//
// VGPR layouts per CDNA5 ISA 7.12.2:
//   A 16x4 : lanes 0-15 rows M=0..15 {K=k0,k0+1}; lanes 16-31 same rows
//            {K=k0+2,k0+3}  (2 VGPRs)
//   B 4x16 : lanes 0-15 cols N=0..15 {K=k0,k0+1}; lanes 16-31 {K=k0+2,k0+3}
//   C 16x16: VGPR r -> row r + 8*(lane>=16), col = lane&15  (8 VGPRs)
// ---------------------------------------------------------------------------
template <int N, int K>
__global__ __launch_bounds__(256) void wmma_gemm_f32(
    const float* __restrict__ A, const float* __restrict__ B,
    float* __restrict__ C, int M, int strideA, int strideB, int strideC)
{
    A += blockIdx.y * strideA;
    B += blockIdx.y * strideB;
    C += blockIdx.y * strideC;

    const int lane = threadIdx.x & 31;
    const int wave = threadIdx.x >> 5;
    const int tile = blockIdx.x * 8 + wave;

    constexpr int ntilesN = N >> 4;                 // power of two
    const int tm = (tile / ntilesN) << 4;           // folds to shift
    const int tn = (tile % ntilesN) << 4;           // folds to mask
    if (tm >= M) return;          // uniform per wave -> EXEC stays all-1s

    const int half = lane >> 4;   // 0: K=k0..k0+1, 1: K=k0+2..k0+3
    const int l    = lane & 15;

    const float* Arow = A + (tm + l) * K + 2 * half;
    const float* Bcol = B + (tn + l) + 2 * half * N;

    v8f acc = {};
#pragma unroll 4
    for (int k0 = 0; k0 < K; k0 += 4) {
        v2f a = *(const v2f*)(Arow + k0);   // A[tm+l][kk..kk+1], 8B aligned
        v2f b;
        b.x = Bcol[k0 * N];                 // B[kk  ][tn+l]
        b.y = Bcol[k0 * N + N];             // B[kk+1][tn+l]
        acc = __builtin_amdgcn_wmma_f32_16x16x4_f32(
            false, a, false, b, (short)0, acc, false, false);
    }

    float* Cbase = C + (tm + 8 * half) * N + (tn + l);
#pragma unroll
    for (int r = 0; r < 8; ++r)
        Cbase[r * N] = acc[r];
}

// ---------------------------------------------------------------------------
// Fused additive-attention energies + softmax, tiled over decoder positions.
// One block (256 threads = 8 waves) handles (b, 8 consecutive s values):
//   alpha[si][t] = sum_e v[e] * tanh(encW[b,t,e] + decW[b,s0+si,e])
//   weights[b,s0+si,:] = softmax_t(alpha[si][:])
//
// encW[b] is streamed through LDS in 16-row chunks (double-buffered) via
// CDNA5 async global->LDS loads; each staged element feeds 8 tanh+fma
// (one per si), making the kernel TRANS-bound instead of L2-bound.
// decW rows and v live in registers (dd[8][8], vv[8] per lane).
// ---------------------------------------------------------------------------
#define S_TILE  8
#define T_CHUNK 16
#define NCHUNK  (T_ENC / T_CHUNK)

__global__ __launch_bounds__(256) void attn_alpha_softmax(
    const float* __restrict__ encW,   // (B, T_ENC, D_ENC)
    const float* __restrict__ decW,   // (B, T_DEC, D_ENC)
    const float* __restrict__ v,      // (D_ENC)
    float* __restrict__ weights)      // (B, T_DEC, T_ENC)
{
    const int b    = blockIdx.y;
    const int s0   = blockIdx.x * S_TILE;
    const int tid  = threadIdx.x;
    const int lane = tid & 31;
    const int wave = tid >> 5;

    __shared__ float lds_enc[2][T_CHUNK * D_ENC];   // 2 x 16KB
    __shared__ float s_alpha[S_TILE][T_ENC];        // 16KB

    const float* encb = encW + b * (T_ENC * D_ENC);

    // per-lane caches: v and the 8 decoder rows (e = j*32 + lane)
    float vv[8], dd[S_TILE][8];
#pragma unroll
    for (int j = 0; j < 8; ++j) vv[j] = v[j * 32 + lane];
#pragma unroll
    for (int si = 0; si < S_TILE; ++si) {
        const float* dec_s = decW + (b * T_DEC + s0 + si) * D_ENC;
#pragma unroll
        for (int j = 0; j < 8; ++j) dd[si][j] = dec_s[j * 32 + lane];
    }

    // stage chunk c: 16 rows = 16KB = 4 x b128 per thread
    auto stage = [&](int buf, int c) {
        const float* src = encb + c * (T_CHUNK * D_ENC);
        float* dst = lds_enc[buf];
#pragma unroll
        for (int i = 0; i < 4; ++i) {
            const int off = tid * 4 + i * 1024;
            copy16_to_lds(src + off, dst + off);
        }
    };

    stage(0, 0);
    wait_async_lds();
    __syncthreads();

    for (int c = 0; c < NCHUNK; ++c) {
        const int buf = c & 1;
        if (c + 1 < NCHUNK) stage(buf ^ 1, c + 1);

        // this wave owns 2 of the 16 staged rows
#pragma unroll
        for (int r = 0; r < 2; ++r) {
            const int tl = wave * 2 + r;
            const float* er_p = &lds_enc[buf][tl * D_ENC];
            float er[8];
#pragma unroll
            for (int j = 0; j < 8; ++j) er[j] = er_p[j * 32 + lane];

#pragma unroll
            for (int si = 0; si < S_TILE; ++si) {
                float acc = 0.f;
#pragma unroll
                for (int j = 0; j < 8; ++j)
                    acc = fmaf(vv[j], fast_tanh(er[j] + dd[si][j]), acc);
#pragma unroll
                for (int off = 16; off > 0; off >>= 1)
                    acc += __shfl_xor(acc, off, 32);
                if (lane == 0) s_alpha[si][c * T_CHUNK + tl] = acc;
            }
        }

        wait_async_lds();   // next chunk landed (my wave's issues)
        __syncthreads();    // everyone done reading buf + next chunk visible
    }

    // ---- softmax: wave w owns row si = w (512 entries, 16 per lane) ----
    const int si = wave;
    float loc[16];
#pragma unroll
    for (int i = 0; i < 16; ++i) loc[i] = s_alpha[si][i * 32 + lane];

    float m = loc[0];
#pragma unroll
    for (int i = 1; i < 16; ++i) m = fmaxf(m, loc[i]);
#pragma unroll
    for (int off = 16; off > 0; off >>= 1)
        m = fmaxf(m, __shfl_xor(m, off, 32));

    float sum = 0.f;
#pragma unroll
    for (int i = 0; i < 16; ++i) {
        loc[i] = expf(loc[i] - m);
        sum += loc[i];
    }
#pragma unroll
    for (int off = 16; off > 0; off >>= 1)
        sum += __shfl_xor(sum, off, 32);
    const float inv = 1.f / sum;

    float* wrow = weights + (b * T_DEC + s0 + si) * T_ENC;
#pragma unroll
    for (int i = 0; i < 16; ++i)
        wrow[i * 32 + lane] = loc[i] * inv;
}

// ---------------------------------------------------------------------------
extern "C" void kernel_launch(void* const* d_in, const int* in_sizes, int n_in,
                              void* d_out, int out_size, void* d_ws, size_t ws_size,
                              hipStream_t stream) {
    const float* enc   = (const float*)d_in[0];  // (8, 512, 256)
    const float* dec   = (const float*)d_in[1];  // (8, 256, 512)
    const float* W_enc = (const float*)d_in[2];  // (256, 256)
    const float* W_dec = (const float*)d_in[3];  // (512, 256)
    const float* v     = (const float*)d_in[4];  // (256, 1)

    float* context = (float*)d_out;                               // (8,256,256)
    float* weights = (float*)d_out + BATCH * T_DEC * D_ENC;       // (8,256,512)

    float* encW = (float*)d_ws;                                   // 4 MB
    float* decW = encW + BATCH * T_ENC * D_ENC;                   // 2 MB

    // 1) encW = enc(4096x256) @ W_enc(256x256)
    wmma_gemm_f32<D_ENC, D_ENC><<<dim3((BATCH * T_ENC / 16) * (D_ENC / 16) / 8, 1),
                                  256, 0, stream>>>(
        enc, W_enc, encW, BATCH * T_ENC, 0, 0, 0);

    // 2) decW = dec(2048x512) @ W_dec(512x256)
    wmma_gemm_f32<D_ENC, D_DEC><<<dim3((BATCH * T_DEC / 16) * (D_ENC / 16) / 8, 1),
                                  256, 0, stream>>>(
        dec, W_dec, decW, BATCH * T_DEC, 0, 0, 0);

    // 3) weights = softmax_t( v . tanh(encW[t] + decW[s]) )
    attn_alpha_softmax<<<dim3(T_DEC / S_TILE, BATCH), 256, 0, stream>>>(
        encW, decW, v, weights);

    // 4) context[b] = weights[b](256x512) @ enc[b](512x256)   (batched WMMA)
    wmma_gemm_f32<D_ENC, T_ENC><<<dim3((T_DEC / 16) * (D_ENC / 16) / 8, BATCH),
                                  256, 0, stream>>>(
        weights, enc, context, T_DEC,
        T_DEC * T_ENC, T_ENC * D_ENC, T_DEC * D_ENC);
}